// NCNPredictor_35270271435515
// MI455X (gfx1250) — compile-verified
//
#include <hip/hip_runtime.h>

// N=200000 nodes, D=32 neighbors, E=65536 edges, F=128 features, OUT=1
constexpr int FEAT = 128;
constexpr int DEG  = 32;

typedef __attribute__((ext_vector_type(2))) float v2f;
typedef __attribute__((ext_vector_type(8))) float v8f;

// ---------------------------------------------------------------------------
// Kernel 1: y[n] = dot(x[n,:], W[128:256])  (per-node scalar, one wave/node)
// Each lane loads 16 contiguous bytes -> wave reads one full 512B row.
// ---------------------------------------------------------------------------
__global__ void ncn_node_phi(const float* __restrict__ x,
                             const float* __restrict__ W,
                             float* __restrict__ y, int N) {
  const int wid  = (int)((blockIdx.x * blockDim.x + threadIdx.x) >> 5);
  const int lane = (int)(threadIdx.x & 31);
  if (wid >= N) return;
  const float4 xv = *(const float4*)(x + (size_t)wid * FEAT + lane * 4);
  const float4 wv = *(const float4*)(W + FEAT + lane * 4);
  float s = xv.x * wv.x + xv.y * wv.y + xv.z * wv.z + xv.w * wv.w;
#pragma unroll
  for (int off = 16; off >= 1; off >>= 1) s += __shfl_xor(s, off, 32);
  if (lane == 0) y[wid] = s;
}

// ---------------------------------------------------------------------------
// Kernel 2: one wave32 handles 16 edges.
//  Stage 1: sorted neighbor-list intersection (32 lanes = 32 slots),
//           branchless binary search via ds_bpermute shuffles, accumulate
//           s_e = sum over common neighbors of y[n]  (algebraic reduction of
//           the reference's spmm_add + W_hi dot, valid because OUT==1).
//  Stage 2: batched weighted dot  d_e = sum_k xi[k]*xj[k]*w[k]  for the 16
//           edges via V_WMMA_F32_16X16X4_F32 (B column 0 = w chunk,
//           selected branchlessly so the hot loop has no exec-mask traffic).
//  Stage 3: out[e] = d_e + s_e + bias.
// ---------------------------------------------------------------------------
__global__ void ncn_edge_kernel(const float* __restrict__ x,
                                const int*  __restrict__ nbr,
                                const int*  __restrict__ tar_ei,
                                const float* __restrict__ W,
                                const float* __restrict__ bptr,
                                const float* __restrict__ y,
                                float* __restrict__ out, int E) {
  const int lane  = (int)(threadIdx.x & 31);
  const int wid   = (int)((blockIdx.x * blockDim.x + threadIdx.x) >> 5);
  const int eBase = wid * 16;
  if (eBase >= E) return;                 // uniform per wave: whole wave exits

  const int m    = lane & 15;             // edge slot within the wave
  const int koff = (lane >> 4) * 2;       // K sub-offset per WMMA A/B layout

  int eIdx = eBase + m;
  if (eIdx > E - 1) eIdx = E - 1;         // clamp (E is a multiple of 16 here)
  const int src = tar_ei[eIdx];           // tar_ei[0][e]
  const int dst = tar_ei[E + eIdx];       // tar_ei[1][e]

  // ---------------- Stage 1: common-neighbor scalar sums -------------------
  float sreg[8];
#pragma unroll
  for (int ee = 0; ee < 16; ++ee) {
    const int eSrc = __shfl(src, ee, 32);
    const int eDst = __shfl(dst, ee, 32);
    const int q    = nbr[(size_t)eSrc * DEG + lane];   // ni[lane]
    const int njv  = nbr[(size_t)eDst * DEG + lane];   // nj[lane] (sorted)

    // branchless lower-bound of q in nj (distributed across lanes)
    int pos = 0;
#pragma unroll
    for (int b = 16; b >= 1; b >>= 1) {
      const int v = __shfl(njv, pos + b - 1, 32);
      if (v < q) pos += b;                 // pos = min(count_less, 31)
    }
    const bool hit = (__shfl(njv, pos, 32) == q);

    // hits are ~0.005/edge: keep the exec-branch so loads are skipped
    float sv = 0.0f;
    if (hit) sv = y[q];
#pragma unroll
    for (int off = 16; off >= 1; off >>= 1) sv += __shfl_xor(sv, off, 32);
    // lanes 0..15 keep edges 0..7, lanes 16..31 keep edges 8..15
    if ((lane >> 4) == (ee >> 3)) sreg[ee & 7] = sv;
  }

  // ---------------- Stage 2: WMMA batched weighted dot ---------------------
  // A (16x4 f32): lane L<16 -> M=L, K={k,k+1}; lane L+16 -> M=L, K={k+2,k+3}
  // B (4x16 f32): column 0 = w[k..k+3] -> only lanes 0 and 16 nonzero.
  // All lanes load the same 8B W chunk per half-wave (L0 broadcast hit) and
  // the column-0 mask is applied with v_cndmask, keeping EXEC untouched.
  const float* xi_row = x + (size_t)src * FEAT;
  const float* xj_row = x + (size_t)dst * FEAT;
  const bool col0 = (m == 0);

  v8f c = {0.f, 0.f, 0.f, 0.f, 0.f, 0.f, 0.f, 0.f};
#pragma unroll 8
  for (int k = 0; k < FEAT; k += 4) {
    const v2f xiv = *(const v2f*)(xi_row + k + koff);
    const v2f xjv = *(const v2f*)(xj_row + k + koff);
    const v2f wv  = *(const v2f*)(W + k + koff);       // uniform per half-wave
    const v2f a   = xiv * xjv;                         // element-wise xi*xj
    v2f bv;
    bv.x = col0 ? wv.x : 0.0f;                         // branchless column-0
    bv.y = col0 ? wv.y : 0.0f;
    c = __builtin_amdgcn_wmma_f32_16x16x4_f32(
            /*neg_a=*/false, a, /*neg_b=*/false, bv,
            /*c_mod=*/(short)0, c, /*reuse_a=*/false, /*reuse_b=*/false);
  }

  // ---------------- Stage 3: write out -------------------------------------
  // D column 0: edges 0..7 -> lane 0, c[0..7]; edges 8..15 -> lane 16, c[0..7]
  const float bias = bptr[0];
  if (lane == 0) {
#pragma unroll
    for (int r = 0; r < 8; ++r) {
      const int e = eBase + r;
      if (e < E) out[e] = c[r] + sreg[r] + bias;
    }
  } else if (lane == 16) {
#pragma unroll
    for (int r = 0; r < 8; ++r) {
      const int e = eBase + 8 + r;
      if (e < E) out[e] = c[r] + sreg[r] + bias;
    }
  }
}

// ---------------------------------------------------------------------------
extern "C" void kernel_launch(void* const* d_in, const int* in_sizes, int n_in,
                              void* d_out, int out_size, void* d_ws, size_t ws_size,
                              hipStream_t stream) {
  const float* x      = (const float*)d_in[0];   // [N,128] f32
  const int*   nbr    = (const int*)  d_in[1];   // [N,32]  i32 (sorted rows)
  const int*   tar_ei = (const int*)  d_in[2];   // [2,E]   i32
  const float* W      = (const float*)d_in[3];   // [256,1] f32
  const float* b      = (const float*)d_in[4];   // [1]     f32
  float*       out    = (float*)d_out;           // [E,1]   f32

  const int N = in_sizes[0] / FEAT;
  const int E = in_sizes[2] / 2;

  float* y = (float*)d_ws;                       // N floats of scratch

  const int threads = 256;                       // 8 waves per block
  // Kernel 1: one wave per node
  const int blocks1 = (int)(((long long)N * 32 + threads - 1) / threads);
  ncn_node_phi<<<blocks1, threads, 0, stream>>>(x, W, y, N);

  // Kernel 2: 16 edges per wave -> 128 edges per block
  const int edgesPerBlock = (threads / 32) * 16;
  const int blocks2 = (E + edgesPerBlock - 1) / edgesPerBlock;
  ncn_edge_kernel<<<blocks2, threads, 0, stream>>>(x, nbr, tar_ei, W, b, y, out, E);
}